// dca_layer_35596688949724
// MI455X (gfx1250) — compile-verified
//
#include <hip/hip_runtime.h>
#include <math.h>

typedef float v2f __attribute__((ext_vector_type(2)));
typedef float v4f __attribute__((ext_vector_type(4)));
typedef float v8f __attribute__((ext_vector_type(8)));

#define B_SZ   16
#define C_SZ   2048
#define HW     3136            // 56*56
#define NF4    784             // HW/4
#define PLANES (B_SZ * C_SZ)   // 32768

// ---------------------------------------------------------------------------
// Kernel 1: global average pool.  One block = 512 threads = 16 waves,
// one wave per (b,c) plane.  Coalesced v4f loads (regular temporal hint so
// the tail of x survives in the 192MB L2 for kernel 3's reverse walk).
// Final 16x32 partial matrix is collapsed with chained
// V_WMMA_F32_16X16X4_F32 (B = ones) -> 16 fp32 row sums, no precision loss.
// ---------------------------------------------------------------------------
__global__ __launch_bounds__(512) void gap_kernel(const float* __restrict__ x,
                                                  float* __restrict__ y) {
    __shared__ float part[16][32];
    const int wave = threadIdx.x >> 5;
    const int lane = threadIdx.x & 31;
    const long long plane = (long long)blockIdx.x * 16 + wave;

    const v4f* xp = (const v4f*)(x + plane * (long long)HW);
    float s = 0.0f;
    for (int i = lane; i < NF4; i += 32) {
        v4f v = xp[i];
        s += (v.x + v.y) + (v.z + v.w);
    }
    part[wave][lane] = s;
    __syncthreads();

    if (wave == 0) {                       // wave-uniform branch: EXEC all 1s
        const int row   = lane & 15;       // A-matrix M index for this lane
        const int khalf = (lane >> 4) << 1;// lanes 0-15 hold K0/K1, 16-31 K2/K3
        v8f acc = {};
        v2f bones = {1.0f, 1.0f};          // B = ones(4x16): D row m = sum_k A[m][k]
        #pragma unroll
        for (int kk = 0; kk < 8; ++kk) {   // 8 chunks of K=4 cover 32 partials
            v2f a;
            a.x = part[row][kk * 4 + khalf];
            a.y = part[row][kk * 4 + khalf + 1];
            acc = __builtin_amdgcn_wmma_f32_16x16x4_f32(
                false, a, false, bones, (short)0, acc, false, false);
        }
        // D layout: VGPR r @ lane 0 -> D[r][0]; VGPR r @ lane 16 -> D[r+8][0]
        const float inv = 1.0f / (float)HW;
        const long long cbase = (long long)blockIdx.x * 16;
        if (lane == 0) {
            #pragma unroll
            for (int m = 0; m < 8; ++m) y[cbase + m] = acc[m] * inv;
        } else if (lane == 16) {
            #pragma unroll
            for (int m = 0; m < 8; ++m) y[cbase + 8 + m] = acc[m] * inv;
        }
    }
}

// ---------------------------------------------------------------------------
// Kernel 2: offset conv1d + deformable gather + sigmoid gate.  Tiny (32K
// outputs).  One block per batch row; y row staged in LDS for the arbitrary
// channel gathers.
// ---------------------------------------------------------------------------
__global__ __launch_bounds__(256) void gate_kernel(const float* __restrict__ y,
                                                   const float* __restrict__ w_offset,
                                                   const float* __restrict__ w_deform,
                                                   const float* __restrict__ b_deform,
                                                   float* __restrict__ gate) {
    __shared__ float ys[C_SZ];
    const int b = blockIdx.x;
    const float* yrow = y + (long long)b * C_SZ;
    for (int i = threadIdx.x; i < C_SZ; i += 256) ys[i] = yrow[i];
    __syncthreads();

    float wo[3][3], wd[3];
    #pragma unroll
    for (int k = 0; k < 3; ++k) {
        #pragma unroll
        for (int j = 0; j < 3; ++j) wo[k][j] = w_offset[k * 3 + j];
        wd[k] = w_deform[k];
    }
    const float bd = b_deform[0];

    for (int c = threadIdx.x; c < C_SZ; c += 256) {
        const float ym1 = (c > 0)        ? ys[c - 1] : 0.0f;   // zero padding
        const float y0  = ys[c];
        const float yp1 = (c < C_SZ - 1) ? ys[c + 1] : 0.0f;
        float acc = bd;
        #pragma unroll
        for (int k = 0; k < 3; ++k) {
            // cross-correlation, kernel width 3, pad 1 (lax.conv semantics)
            const float off = wo[k][0] * ym1 + wo[k][1] * y0 + wo[k][2] * yp1;
            float p = (float)c + (float)(k - 1) + off;
            p = fminf(fmaxf(p, 0.0f), (float)(C_SZ - 1));
            const float p0   = floorf(p);
            const float frac = p - p0;
            const int   p0i  = (int)p0;
            const int   p1i  = min(p0i + 1, C_SZ - 1);
            const float v = ys[p0i] * (1.0f - frac) + ys[p1i] * frac;
            acc += v * wd[k];
        }
        gate[(long long)b * C_SZ + c] = 1.0f / (1.0f + expf(-acc));
    }
}

// ---------------------------------------------------------------------------
// Kernel 3: out = x * gate[plane].  Dominant cost (822 MB).  Planes walked in
// REVERSE so the last ~192MB of x cached in L2 by kernel 1 is re-hit first.
// x loads NT (last use), out stores NT (write-once, never re-read on device).
// ---------------------------------------------------------------------------
__global__ __launch_bounds__(256) void scale_kernel(const float* __restrict__ x,
                                                    const float* __restrict__ gate,
                                                    float* __restrict__ out) {
    const int plane = (PLANES - 1) - blockIdx.x;
    const float g = gate[plane];
    const long long base = (long long)plane * (long long)HW;
    const v4f* xp = (const v4f*)(x + base);
    v4f* op = (v4f*)(out + base);
    for (int i = threadIdx.x; i < NF4; i += 256) {
        v4f v = __builtin_nontemporal_load(&xp[i]);
        v.x *= g; v.y *= g; v.z *= g; v.w *= g;
        __builtin_nontemporal_store(v, &op[i]);
    }
}

extern "C" void kernel_launch(void* const* d_in, const int* in_sizes, int n_in,
                              void* d_out, int out_size, void* d_ws, size_t ws_size,
                              hipStream_t stream) {
    (void)in_sizes; (void)n_in; (void)out_size; (void)ws_size;
    const float* x        = (const float*)d_in[0];   // (16,2048,56,56)
    const float* w_offset = (const float*)d_in[1];   // (3,1,3)
    const float* w_deform = (const float*)d_in[2];   // (3,)
    const float* b_deform = (const float*)d_in[3];   // (1,)
    float* out  = (float*)d_out;
    float* y    = (float*)d_ws;                      // 16*2048 floats
    float* gate = y + PLANES;                        // 16*2048 floats

    gap_kernel  <<<PLANES / 16, 512, 0, stream>>>(x, y);
    gate_kernel <<<B_SZ,        256, 0, stream>>>(y, w_offset, w_deform, b_deform, gate);
    scale_kernel<<<PLANES,      256, 0, stream>>>(x, gate, out);
}